// graph_loss_29274497089622
// MI455X (gfx1250) — compile-verified
//
#include <hip/hip_runtime.h>
#include <math.h>

// ---------------------------------------------------------------------------
// Problem constants (from reference): B=2048, D=4096, L=3, 3D=12288
// ---------------------------------------------------------------------------
#define DDIM   4096
#define BDIM   2048
#define G3D    12288   // 3*D
#define NLAY   3

typedef float v2f __attribute__((ext_vector_type(2)));
typedef float v8f __attribute__((ext_vector_type(8)));

struct Ptr8 { const float* p[8]; };

// Async copy: 16B global -> LDS, tracked by ASYNCcnt (gfx1250 path).
__device__ __forceinline__ void async_ld_b128(uint32_t lds_off, const float* g) {
  asm volatile("global_load_async_to_lds_b128 %0, %1, off"
               :: "v"(lds_off), "v"(g) : "memory");
}
// Generic pointer to LDS: low 32 bits of the flat address are the LDS offset.
__device__ __forceinline__ uint32_t lds_off_of(const void* p) {
  return (uint32_t)(uintptr_t)p;
}

// ---------------------------------------------------------------------------
// Kernel 1: batch-mean partial sums.
// grid = 8 tensors * 16 column tiles * 8 row chunks = 1024 blocks, 256 threads.
// ---------------------------------------------------------------------------
__global__ void k_mean_partial(Ptr8 in, float* __restrict__ part) {
  int bid = blockIdx.x;
  int t   = bid >> 7;          // tensor 0..7
  int rem = bid & 127;
  int ct  = rem & 15;          // column tile 0..15
  int ch  = rem >> 4;          // row chunk 0..7
  int col = ct * 256 + threadIdx.x;
  const float* __restrict__ src = in.p[t];
  int r0 = ch * 256;
  float acc = 0.0f;
#pragma unroll 4
  for (int r = 0; r < 256; ++r)
    acc += src[(size_t)(r0 + r) * DDIM + col];
  part[((t * 8 + ch) << 12) + col] = acc;
}

// ---------------------------------------------------------------------------
// Kernel 2: reduce partials -> means8 [8][4096] and initialize H [8][4096].
// H rows 0..3 = now_means (streams' node0), rows 4..7 = pre_means (shared).
// ---------------------------------------------------------------------------
__global__ void k_mean_reduce(const float* __restrict__ part,
                              float* __restrict__ means8,
                              float* __restrict__ H) {
  int idx = blockIdx.x * 256 + threadIdx.x;  // t*4096 + d
  int t = idx >> 12;
  int d = idx & 4095;
  float s = 0.0f;
#pragma unroll
  for (int c = 0; c < 8; ++c)
    s += part[((t * 8 + c) << 12) + d];
  float m = s * (1.0f / (float)BDIM);
  means8[idx] = m;
  H[idx]      = m;
}

// ---------------------------------------------------------------------------
// Kernel 3: agg partial.  agg0 = (sum of shared rows of H) @ W[l].
// grid = 16 column tiles * 16 k-chunks = 256 blocks, 256 threads.
// ---------------------------------------------------------------------------
__global__ void k_agg_partial(const float* __restrict__ H,
                              const float* __restrict__ Wl,
                              float* __restrict__ aggPart) {
  __shared__ float usum[256];
  int ct = blockIdx.x & 15;
  int ch = blockIdx.x >> 4;
  int n  = ct * 256 + threadIdx.x;
  int k0 = ch * 256;
  int k  = k0 + threadIdx.x;
  usum[threadIdx.x] = H[(4 << 12) + k] + H[(5 << 12) + k] +
                      H[(6 << 12) + k] + H[(7 << 12) + k];
  __syncthreads();
  float acc = 0.0f;
#pragma unroll 4
  for (int kk = 0; kk < 256; ++kk)
    acc = fmaf(usum[kk], Wl[(size_t)(k0 + kk) * DDIM + n], acc);
  aggPart[(ch << 12) + n] = acc;
}

// ---------------------------------------------------------------------------
// Kernel 4: reduce agg partials -> agg0 [4096]. grid = 16 * 256.
// ---------------------------------------------------------------------------
__global__ void k_agg_reduce(const float* __restrict__ aggPart,
                             float* __restrict__ agg0) {
  int n = blockIdx.x * 256 + threadIdx.x;
  float s = 0.0f;
#pragma unroll
  for (int c = 0; c < 16; ++c)
    s += aggPart[(c << 12) + n];
  agg0[n] = s;
}

// ---------------------------------------------------------------------------
// Kernel 5: WMMA GEMM with double-buffered async global->LDS staging.
//   Out[m][j] = sum_k A[m][k] * Wt[j][k]
//   A   : [rowsA <= 16][4096] (rows >= rowsA are loaded but their D rows are
//         never stored, so no zero-fill needed -> wave-uniform ASYNCcnt)
//   Wt  : [12288][4096] row-major
//   Out : 2-way K-split partials; OutPart[ks*outKsStride + m*12288 + j]
// grid = 192 j-tiles * 2 k-splits = 384 blocks, 128 threads (4 waves).
// Pipeline per 64-wide K chunk (10 async b128 ops per thread, uniform):
//   issue chunk i+1 -> s_wait_asynccnt 0xa (chunk i complete) -> barrier ->
//   16x v_wmma_f32_16x16x4_f32 from LDS -> barrier (protect re-use of buffer)
// ---------------------------------------------------------------------------
#define KC 64
__global__ void k_gemm_wmma(const float* __restrict__ A,
                            const float* __restrict__ Wt,
                            float* __restrict__ OutPart,
                            int rowsA, int outKsStride) {
  __shared__ float ldsB[2][64 * KC];   // [buf][j_local][k_local] 16KB each
  __shared__ float ldsA[2][16 * KC];   // [buf][m][k_local]        4KB each
  int bid  = blockIdx.x;
  int jt   = (bid % 192) * 64;
  int ks   = bid / 192;              // 0 or 1
  int kBeg = ks * 2048;
  int lane = threadIdx.x & 31;
  int wave = threadIdx.x >> 5;
  int mn   = lane & 15;              // M (A frag) / N (B frag) within tile
  int half = lane >> 4;              // selects K pair {0,1} vs {2,3}

  v8f acc = {0.0f, 0.0f, 0.0f, 0.0f, 0.0f, 0.0f, 0.0f, 0.0f};

  // Issue one chunk's async copies: 8 B-tile ops + 2 A-tile ops per thread.
  auto issue_chunk = [&](int k0, int buf) {
    for (int i = threadIdx.x; i < 64 * (KC / 4); i += 128) {   // 8 iters/thread
      int r = i >> 4;                 // j row 0..63
      int c = (i & 15) << 2;          // k col, step 4 floats (16B)
      async_ld_b128(lds_off_of(&ldsB[buf][r * KC + c]),
                    &Wt[(size_t)(jt + r) * DDIM + k0 + c]);
    }
    for (int i = threadIdx.x; i < 16 * (KC / 4); i += 128) {   // 2 iters/thread
      int r = i >> 4;                 // m row 0..15
      int c = (i & 15) << 2;
      async_ld_b128(lds_off_of(&ldsA[buf][r * KC + c]),
                    &A[(size_t)r * DDIM + k0 + c]);
    }
  };

  const int nch = 2048 / KC;          // 32 chunks
  issue_chunk(kBeg, 0);
  for (int ci = 0; ci < nch; ++ci) {
    int buf = ci & 1;
    if (ci + 1 < nch) {
      issue_chunk(kBeg + (ci + 1) * KC, buf ^ 1);
      // keep only the 10 in-flight ops of chunk ci+1 -> chunk ci is complete
      asm volatile("s_wait_asynccnt 0xa" ::: "memory");
    } else {
      asm volatile("s_wait_asynccnt 0x0" ::: "memory");
    }
    if (ci + 4 < nch)  // far prefetch of the B stream (global_prefetch_b8)
      __builtin_prefetch(&Wt[(size_t)(jt + (threadIdx.x >> 1)) * DDIM +
                             kBeg + (ci + 4) * KC], 0, 0);
    __syncthreads();   // every wave waited its own ASYNCcnt -> tile visible

    const float* pa = &ldsA[buf][mn * KC + 2 * half];
    const float* pb = &ldsB[buf][(wave * 16 + mn) * KC + 2 * half];
#pragma unroll
    for (int kk = 0; kk < KC; kk += 4) {
      // A frag (16x4 f32): lane m=L%16, VGPR0=K(2*half), VGPR1=K(2*half+1)
      v2f a = *(const v2f*)&pa[kk];
      // B frag (4x16 f32): lane n=L%16, mirrored K layout
      v2f b = *(const v2f*)&pb[kk];
      acc = __builtin_amdgcn_wmma_f32_16x16x4_f32(
          /*neg_a=*/false, a, /*neg_b=*/false, b,
          /*c_mod=*/(short)0, acc, /*reuse_a=*/false, /*reuse_b=*/false);
    }
    __syncthreads();   // all waves done reading buf before it is re-filled
  }

  // D layout: VGPR i -> lanes 0-15: M=i, N=lane; lanes 16-31: M=i+8, N=lane-16
  int j = jt + wave * 16 + mn;
#pragma unroll
  for (int i = 0; i < 8; ++i) {
    int m = i + half * 8;
    if (m < rowsA)
      OutPart[(size_t)ks * outKsStride + (size_t)m * G3D + j] = acc[i];
  }
}

// ---------------------------------------------------------------------------
// Kernel 6: fused GRU update (torch GRUCell semantics), in-place on H.
// Sums the 2-way K-split partials of gi0 / gh here.
// ---------------------------------------------------------------------------
__global__ void k_gru(float* __restrict__ H,
                      const float* __restrict__ gi0p,   // [2][12288]
                      const float* __restrict__ ghp,    // [2][8][12288]
                      const float* __restrict__ b_ih,
                      const float* __restrict__ b_hh) {
  int idx = blockIdx.x * 256 + threadIdx.x;  // m*4096 + d
  int m = idx >> 12;
  int d = idx & 4095;

  float ir = b_ih[d];
  float iz = b_ih[DDIM + d];
  float in_ = b_ih[2 * DDIM + d];
  if (m < 4) {  // stream rows receive the (stream-invariant) message gi0
    ir  += gi0p[d]            + gi0p[G3D + d];
    iz  += gi0p[DDIM + d]     + gi0p[G3D + DDIM + d];
    in_ += gi0p[2 * DDIM + d] + gi0p[G3D + 2 * DDIM + d];
  }
  size_t gb = (size_t)m * G3D;
  const size_t KS = (size_t)8 * G3D;  // 98304
  float hr = ghp[gb + d]            + ghp[KS + gb + d]            + b_hh[d];
  float hz = ghp[gb + DDIM + d]     + ghp[KS + gb + DDIM + d]     + b_hh[DDIM + d];
  float hn = ghp[gb + 2 * DDIM + d] + ghp[KS + gb + 2 * DDIM + d] + b_hh[2 * DDIM + d];

  float r = 1.0f / (1.0f + expf(-(ir + hr)));
  float z = 1.0f / (1.0f + expf(-(iz + hz)));
  float n = tanhf(in_ + r * hn);
  float h = H[idx];
  H[idx] = (1.0f - z) * n + z * h;
}

// ---------------------------------------------------------------------------
// Kernel 7: final loss = 10 * sum_s mean_d (H[s] - pre_means[s])^2
// ---------------------------------------------------------------------------
__global__ void k_loss(const float* __restrict__ H,
                       const float* __restrict__ means8,
                       float* __restrict__ out) {
  __shared__ float red[256];
  float acc = 0.0f;
  for (int idx = threadIdx.x; idx < 4 * DDIM; idx += 256) {
    int s = idx >> 12;
    int d = idx & 4095;
    float diff = H[(s << 12) + d] - means8[((s + 4) << 12) + d];
    acc += diff * diff;
  }
  red[threadIdx.x] = acc;
  __syncthreads();
  for (int off = 128; off > 0; off >>= 1) {
    if ((int)threadIdx.x < off) red[threadIdx.x] += red[threadIdx.x + off];
    __syncthreads();
  }
  if (threadIdx.x == 0) out[0] = red[0] * (10.0f / (float)DDIM);
}

// ---------------------------------------------------------------------------
// Host launcher
// ---------------------------------------------------------------------------
extern "C" void kernel_launch(void* const* d_in, const int* in_sizes, int n_in,
                              void* d_out, int out_size, void* d_ws, size_t ws_size,
                              hipStream_t stream) {
  const float* W    = (const float*)d_in[8];   // [3][4096][4096]
  const float* w_ih = (const float*)d_in[9];   // [12288][4096]
  const float* w_hh = (const float*)d_in[10];  // [12288][4096]
  const float* b_ih = (const float*)d_in[11];  // [12288]
  const float* b_hh = (const float*)d_in[12];  // [12288]

  float* ws        = (float*)d_ws;
  float* meansPart = ws;                       // 8*8*4096   = 262144
  float* means8    = meansPart + 262144;       // 8*4096     =  32768
  float* H         = means8    + 32768;        // 8*4096     =  32768
  float* aggPart   = H         + 32768;        // 16*4096    =  65536
  float* agg0      = aggPart   + 65536;        // 4096
  float* gi0p      = agg0      + 4096;         // 2*12288    =  24576
  float* ghp       = gi0p      + 24576;        // 2*8*12288  = 196608
  // total ~2.4 MB of workspace

  Ptr8 ptrs;
  for (int i = 0; i < 8; ++i) ptrs.p[i] = (const float*)d_in[i];

  // Phase 1: batch means (268 MB streaming)
  k_mean_partial<<<1024, 256, 0, stream>>>(ptrs, meansPart);
  k_mean_reduce <<<128,  256, 0, stream>>>(meansPart, means8, H);

  // Phase 2: 3 GatedGraphConv+GRU layers (~470 MB weight streaming each)
  for (int l = 0; l < NLAY; ++l) {
    const float* Wl = W + (size_t)l * DDIM * DDIM;
    k_agg_partial<<<256, 256, 0, stream>>>(H, Wl, aggPart);
    k_agg_reduce <<<16,  256, 0, stream>>>(aggPart, agg0);
    // gi0 = agg0 @ w_ih^T   (M=1 padded into 16-row WMMA tile)
    k_gemm_wmma<<<384, 128, 0, stream>>>(agg0, w_ih, gi0p, 1, G3D);
    // gh  = H @ w_hh^T      (M=8 padded into 16-row WMMA tile)
    k_gemm_wmma<<<384, 128, 0, stream>>>(H, w_hh, ghp, 8, 8 * G3D);
    k_gru<<<128, 256, 0, stream>>>(H, gi0p, ghp, b_ih, b_hh);
  }

  // Phase 3: scalar loss
  k_loss<<<1, 256, 0, stream>>>(H, means8, (float*)d_out);
}